// E3ConvLayer_78958678769811
// MI455X (gfx1250) — compile-verified
//
#include <hip/hip_runtime.h>

typedef __attribute__((ext_vector_type(2))) float v2f;
typedef __attribute__((ext_vector_type(8))) float v8f;

#define NBR    41
#define SHDIM  9
#define MDEG   12
#define FDIM   64
#define KSTEPS 11    // K padded 41 -> 44 (row 41 = bias, rows 42/43 = 0)
#define HPAD   48    // hidden cols padded 41 -> 48 (3 x 16)
#define NTILE  32    // nodes per block
#define ETILES 24    // 32*12/16 edge tiles per block

__device__ __forceinline__ float softplus_f(float x) {
    // numerically stable: max(x,0) + log1p(exp(-|x|))
    return fmaxf(x, 0.0f) + log1pf(__expf(-fabsf(x)));
}

__global__ __launch_bounds__(256)
void e3conv_fused_kernel(const float* __restrict__ atom_fea,
                         const float* __restrict__ nbr_fea,
                         const int*   __restrict__ nbr_idx,
                         const float* __restrict__ w1,
                         const float* __restrict__ b1,
                         const float* __restrict__ w2,
                         const float* __restrict__ b2,
                         const float* __restrict__ tp_w,
                         float* __restrict__ out,
                         int n_nodes)
{
    // ---- LDS (~35 KB) ----
    __shared__ v2f   Bs2[KSTEPS][3][32];    // radial-GEMM B fragments (w1 + bias row)
    __shared__ v2f   Tw[16][4][32];         // output-GEMM B fragments (tp_w)
    __shared__ float w2c0[HPAD];            // column 0 of w2, zero-padded
    __shared__ float alphaS[NTILE * MDEG];  // per-edge scalar weight
    __shared__ float gs[NTILE][FDIM + 4];   // gathered/weighted features (A for phase C)

    const int tid  = threadIdx.x;
    const int lane = tid & 31;
    const int wave = tid >> 5;              // 0..7
    const int node0 = blockIdx.x * NTILE;

    // -------- stage shared operands --------
    for (int i = tid; i < HPAD; i += 256)
        w2c0[i] = (i < NBR) ? w2[i * SHDIM + 0] : 0.0f;

    // Radial GEMM B: B[r][c], r in [0,44): r<41 -> w1[r][c]; r==41 -> b1[c]; else 0.
    // Fragment layout: slot<16 holds rows {4s, 4s+1}, slot>=16 holds rows {4s+2, 4s+3},
    // column = 16*t + slot%16 (zero for c >= 41).
    for (int s = tid; s < KSTEPS * 3 * 32; s += 256) {
        int slot = s & 31;
        int t    = (s >> 5) % 3;
        int st   = s / (3 * 32);
        int r = 4 * st + ((slot < 16) ? 0 : 2);
        int c = 16 * t + (slot & 15);
        float x0 = 0.0f, x1 = 0.0f;
        if (c < NBR) {
            if (r < NBR)           x0 = w1[r * NBR + c];
            else if (r == NBR)     x0 = b1[c];
            int r1 = r + 1;
            if (r1 < NBR)          x1 = w1[r1 * NBR + c];
            else if (r1 == NBR)    x1 = b1[c];
        }
        v2f v; v.x = x0; v.y = x1;
        Bs2[st][t][slot] = v;
    }

    // Output GEMM B: tp_w[k][n], k,n in [0,64)
    for (int s = tid; s < 16 * 4 * 32; s += 256) {
        int slot = s & 31;
        int w    = (s >> 5) & 3;
        int st   = s >> 7;
        int r = 4 * st + ((slot < 16) ? 0 : 2);
        int c = 16 * w + (slot & 15);
        v2f v; v.x = tp_w[r * FDIM + c]; v.y = tp_w[(r + 1) * FDIM + c];
        Tw[st][w][slot] = v;
    }
    __syncthreads();

    // -------- Phase A: radial MLP -> alpha per edge (fp32 WMMA) --------
    const float SH_C  = 0.28209479177387814f;            // 1/sqrt(4*pi) == Y_{l=0}
    const float SCALE = SH_C / (8.0f * (float)MDEG);     // / sqrt(F) / M
    const float b2_0  = b2[0];

    const long long emax = (long long)n_nodes * MDEG - 1;
    const int koff = (lane < 16) ? 0 : 2;

    for (int et = wave; et < ETILES; et += 8) {
        const int e0 = node0 * MDEG + et * 16;
        long long erow = (long long)e0 + (lane & 15);
        if (erow > emax) erow = emax;                     // tail-safe loads
        const float* Aptr = nbr_fea + erow * NBR;

        v8f c0 = {}; v8f c1 = {}; v8f c2 = {};
        #pragma unroll
        for (int st = 0; st < KSTEPS; ++st) {
            const int k = 4 * st + koff;
            v2f a;
            a.x = (k     < NBR) ? Aptr[k]     : ((k     == NBR) ? 1.0f : 0.0f);
            a.y = (k + 1 < NBR) ? Aptr[k + 1] : ((k + 1 == NBR) ? 1.0f : 0.0f);
            c0 = __builtin_amdgcn_wmma_f32_16x16x4_f32(false, a, false, Bs2[st][0][lane], (short)0, c0, false, false);
            c1 = __builtin_amdgcn_wmma_f32_16x16x4_f32(false, a, false, Bs2[st][1][lane], (short)0, c1, false, false);
            c2 = __builtin_amdgcn_wmma_f32_16x16x4_f32(false, a, false, Bs2[st][2][lane], (short)0, c2, false, false);
        }

        // softplus, dot with w2[:,0], reduce over hidden cols (16 lanes per half)
        const int n = lane & 15;
        const float wa = w2c0[n], wb = w2c0[16 + n], wc = w2c0[32 + n];
        #pragma unroll
        for (int r = 0; r < 8; ++r) {
            float p = softplus_f(c0[r]) * wa
                    + softplus_f(c1[r]) * wb
                    + softplus_f(c2[r]) * wc;
            p += __shfl_xor(p, 1, 32);
            p += __shfl_xor(p, 2, 32);
            p += __shfl_xor(p, 4, 32);
            p += __shfl_xor(p, 8, 32);
            if ((lane & 15) == 0) {
                int edge_local = et * 16 + r + ((lane >> 4) << 3);  // upper half = rows 8..15
                alphaS[edge_local] = SCALE * (p + b2_0);
            }
        }
    }
    __syncthreads();

    // -------- Phase B: g[nl][f] = sum_m alpha * atom_fea[nbr_idx] --------
    {
        const int f   = tid & 63;
        const int grp = tid >> 6;            // 0..3
        for (int nl = grp; nl < NTILE; nl += 4) {
            int node = node0 + nl;
            int nc = (node < n_nodes) ? node : (n_nodes - 1);
            float acc = 0.0f;
            #pragma unroll
            for (int m = 0; m < MDEG; ++m) {
                int j = nbr_idx[nc * MDEG + m];
                acc = fmaf(alphaS[nl * MDEG + m], atom_fea[(long long)j * FDIM + f], acc);
            }
            gs[nl][f] = acc;
        }
    }
    __syncthreads();

    // -------- Phase C: out tile = g @ tp_w (fp32 WMMA) --------
    // 8 output 16x16 tiles: wave -> (node-row = wave>>2, col-tile = wave&3)
    {
        const int wcol = wave & 3;
        const int nrow = (wave >> 2) * 16;
        const int row  = nrow + (lane & 15);
        v8f acc = {};
        #pragma unroll
        for (int st = 0; st < 16; ++st) {
            const int k = 4 * st + koff;
            v2f a; a.x = gs[row][k]; a.y = gs[row][k + 1];
            acc = __builtin_amdgcn_wmma_f32_16x16x4_f32(false, a, false, Tw[st][wcol][lane], (short)0, acc, false, false);
        }
        const int col   = 16 * wcol + (lane & 15);
        const int rbase = (lane < 16) ? 0 : 8;
        #pragma unroll
        for (int r = 0; r < 8; ++r) {
            int node = node0 + nrow + rbase + r;
            if (node < n_nodes)
                out[(long long)node * FDIM + col] = acc[r];
        }
    }
}

extern "C" void kernel_launch(void* const* d_in, const int* in_sizes, int n_in,
                              void* d_out, int out_size, void* d_ws, size_t ws_size,
                              hipStream_t stream)
{
    const float* atom_fea = (const float*)d_in[0];
    const float* nbr_fea  = (const float*)d_in[1];
    const int*   nbr_idx  = (const int*)d_in[2];
    // d_in[3] = pos: unused — Y_{l=0} of a normalized vector is the constant 1/sqrt(4*pi)
    const float* w1   = (const float*)d_in[4];
    const float* b1   = (const float*)d_in[5];
    const float* w2   = (const float*)d_in[6];
    const float* b2   = (const float*)d_in[7];
    const float* tp_w = (const float*)d_in[8];
    float* out = (float*)d_out;

    const int n_nodes = in_sizes[0] / FDIM;
    const int grid = (n_nodes + NTILE - 1) / NTILE;
    e3conv_fused_kernel<<<grid, 256, 0, stream>>>(atom_fea, nbr_fea, nbr_idx,
                                                  w1, b1, w2, b2, tp_w, out, n_nodes);
}